// UnsupervisedDiffusionLoss_58669253263727
// MI455X (gfx1250) — compile-verified
//
#include <hip/hip_runtime.h>
#include <hip/hip_bf16.h>
#include <math.h>

typedef __attribute__((ext_vector_type(2))) float v2f;
typedef __attribute__((ext_vector_type(8))) float v8f;

#define BATCHES          4
#define N_PTS            8192
#define WAVES_PER_BLOCK  8
#define TILES_PER_WAVE   2                       // 32 rows per wave
#define ROWS_PER_WAVE    (16 * TILES_PER_WAVE)
#define ROWS_PER_BLOCK   (ROWS_PER_WAVE * WAVES_PER_BLOCK)   // 256
#define BLOCKS_PER_BATCH (N_PTS / ROWS_PER_BLOCK)            // 32
#define CHAMFER_GRID     (BATCHES * BLOCKS_PER_BATCH)        // 128 per direction
#define WAVES_PER_DIR    (CHAMFER_GRID * WAVES_PER_BLOCK)    // 1024

#define MSE_BLOCKS  64       // blocks per MSE region
#define MSE_THREADS 256

// Workspace layout (floats):
//   [0,   64)   diff MSE block partials
//   [64, 128)   style MSE block partials
//   [128,192)   content MSE block partials
//   [192,256)   identity MSE block partials
//   [256, 256+1024)       cd1 wave partials (x_original rows vs x_cycle cols)
//   [256+1024, 256+2048)  cd2 wave partials (swapped)

// ---------------------------------------------------------------------------
// Chamfer, both directions in one launch (grid = 2 * CHAMFER_GRID).
// For each row of X, min over all Y of squared distance, then sqrt and sum.
// x.y inner products via V_WMMA_F32_16X16X4_F32 (K=3 padded to 4 with zeros).
// ---------------------------------------------------------------------------
__global__ __launch_bounds__(256)
void chamfer_both_kernel(const float* __restrict__ P,   // x_original
                         const float* __restrict__ Q,   // x_cycle
                         float* __restrict__ partials)  // ws + 256
{
#pragma float_control(precise, off)
    const int dir  = blockIdx.x / CHAMFER_GRID;   // 0: P rows vs Q; 1: swapped
    const int blk  = blockIdx.x % CHAMFER_GRID;
    const float* X = dir ? Q : P;
    const float* Y = dir ? P : Q;

    const int lane  = threadIdx.x & 31;
    const int wave  = threadIdx.x >> 5;
    const int batch = blk / BLOCKS_PER_BATCH;
    const int rowBase = (blk % BLOCKS_PER_BATCH) * ROWS_PER_BLOCK
                        + wave * ROWS_PER_WAVE;

    const float* Xb = X + (size_t)batch * N_PTS * 3;
    const float* Yb = Y + (size_t)batch * N_PTS * 3;

    const bool hi  = (lane >= 16);   // lanes 16-31 carry (k2,k3) = (z, 0)
    const int  lid = lane & 15;

    // A fragments: 16x4 f32; lane lid is matrix row (rowBase + 16t + lid)
    v2f afrag[TILES_PER_WAVE];
    #pragma unroll
    for (int t = 0; t < TILES_PER_WAVE; ++t) {
        const float* p = Xb + 3 * (rowBase + 16 * t + lid);
        float x = p[0], y = p[1], z = p[2];
        afrag[t].x = hi ? z   : x;
        afrag[t].y = hi ? 0.f : y;
    }

    // Running min of (|y|^2 - 2 x.y) per row; |x|^2 added at the end.
    v8f acc[TILES_PER_WAVE];
    #pragma unroll
    for (int t = 0; t < TILES_PER_WAVE; ++t)
        #pragma unroll
        for (int r = 0; r < 8; ++r)
            acc[t][r] = 3.0e38f;

    const v8f czero = {};

    // Process one 16-column B tile given its point (uniform, EXEC all-ones).
    auto process = [&](float px, float py, float pz) {
        float y2 = fmaf(px, px, fmaf(py, py, pz * pz));
        v2f bfrag;
        bfrag.x = hi ? pz  : px;
        bfrag.y = hi ? 0.f : py;
        #pragma unroll
        for (int t = 0; t < TILES_PER_WAVE; ++t) {
            // D = A x B (+0): 16x16 tile of x.y
            v8f d = __builtin_amdgcn_wmma_f32_16x16x4_f32(
                false, afrag[t], false, bfrag, (short)0, czero, false, false);
            #pragma unroll
            for (int r = 0; r < 8; ++r)
                acc[t][r] = __builtin_fminf(acc[t][r], fmaf(d[r], -2.0f, y2));
        }
    };

    // Software pipeline: prefetch next tile's point while processing the
    // current one; last tile peeled so the prefetch never goes out of bounds.
    const float* p0 = Yb + 3 * lid;
    float px = p0[0], py = p0[1], pz = p0[2];
    for (int col0 = 0; col0 < N_PTS - 16; col0 += 16) {
        const float* q = Yb + 3 * (col0 + 16 + lid);
        float nx = q[0], ny = q[1], nz = q[2];
        process(px, py, pz);
        px = nx; py = ny; pz = nz;
    }
    process(px, py, pz);

    // C/D tile layout: VGPR r, lanes 0-15 -> row r, lanes 16-31 -> row r+8.
    // Min across 16 lanes of each half, then sqrt(|x|^2 + min) per row.
    float rowsum = 0.f;
    #pragma unroll
    for (int t = 0; t < TILES_PER_WAVE; ++t) {
        #pragma unroll
        for (int r = 0; r < 8; ++r) {
            float m = acc[t][r];
            #pragma unroll
            for (int s = 1; s < 16; s <<= 1)
                m = __builtin_fminf(m, __shfl_xor(m, s, 32));
            if (lid == 0) {  // lanes 0 and 16 each own one row
                int row = rowBase + 16 * t + r + (hi ? 8 : 0);
                const float* qx = Xb + 3 * row;
                float x2 = fmaf(qx[0], qx[0], fmaf(qx[1], qx[1], qx[2] * qx[2]));
                rowsum += sqrtf(__builtin_fmaxf(x2 + m, 0.f));
            }
        }
    }
    // Sum the two contributing lanes across the wave, write one partial.
    #pragma unroll
    for (int s = 1; s < 32; s <<= 1)
        rowsum += __shfl_xor(rowsum, s, 32);
    if (lane == 0)
        partials[dir * WAVES_PER_DIR + blk * WAVES_PER_BLOCK + wave] = rowsum;
}

// ---------------------------------------------------------------------------
// All four squared-difference reductions in one launch:
// region = blockIdx.x >> 6 (64 blocks each), one partial per block.
// ---------------------------------------------------------------------------
__global__ __launch_bounds__(MSE_THREADS)
void mse4_partial_kernel(const float* __restrict__ a0, const float* __restrict__ b0,
                         const float* __restrict__ a1, const float* __restrict__ b1,
                         const float* __restrict__ a2, const float* __restrict__ b2,
                         const float* __restrict__ a3, const float* __restrict__ b3,
                         float* __restrict__ partials)
{
    const int region = blockIdx.x >> 6;   // uniform per block
    const int rb     = blockIdx.x & 63;
    const float* a; const float* b; int n;
    if      (region == 0) { a = a0; b = b0; n = 4 * 8192 * 3;   }
    else if (region == 1) { a = a1; b = b1; n = 4 * 512;        }
    else if (region == 2) { a = a2; b = b2; n = 4 * 1024 * 256; }
    else                  { a = a3; b = b3; n = 4 * 8192 * 3;   }

    float s = 0.f;
    for (int i = rb * MSE_THREADS + threadIdx.x; i < n;
         i += MSE_BLOCKS * MSE_THREADS) {
        float d = a[i] - b[i];
        s = fmaf(d, d, s);
    }
    __shared__ float sdata[MSE_THREADS / 32];
    #pragma unroll
    for (int off = 16; off > 0; off >>= 1)
        s += __shfl_xor(s, off, 32);
    int lane = threadIdx.x & 31, w = threadIdx.x >> 5;
    if (lane == 0) sdata[w] = s;
    __syncthreads();
    if (threadIdx.x == 0) {
        float v = 0.f;
        #pragma unroll
        for (int i = 0; i < MSE_THREADS / 32; ++i) v += sdata[i];
        partials[region * MSE_BLOCKS + rb] = v;
    }
}

// ---------------------------------------------------------------------------
// Final reduction of all partial regions + loss assembly.
// ---------------------------------------------------------------------------
__global__ __launch_bounds__(256)
void finalize_kernel(const float* __restrict__ ws, float* __restrict__ out)
{
    __shared__ float red[256];
    float sums[6];
    const int t = threadIdx.x;
    #pragma unroll
    for (int k = 0; k < 6; ++k) {
        const int base = (k < 4) ? (k * 64) : (256 + (k - 4) * WAVES_PER_DIR);
        const int len  = (k < 4) ? 64 : WAVES_PER_DIR;
        float s = 0.f;
        for (int i = t; i < len; i += 256) s += ws[base + i];
        red[t] = s;
        __syncthreads();
        for (int off = 128; off > 0; off >>= 1) {
            if (t < off) red[t] += red[t + off];
            __syncthreads();
        }
        sums[k] = red[0];
        __syncthreads();
    }
    if (t == 0) {
        float diff     = sums[0] / (4.f * 8192.f * 3.f);
        float style    = -(sums[1] / (4.f * 512.f));
        float content  = sums[2] / (4.f * 1024.f * 256.f);
        float identity = sums[3] / (4.f * 8192.f * 3.f);
        float cycle    = (sums[4] + sums[5]) / (2.f * 4.f * 8192.f);
        float total = 1.0f * diff + 0.5f * style + 0.5f * content
                    + 1.0f * cycle + 0.5f * identity;
        out[0] = diff; out[1] = style; out[2] = content;
        out[3] = cycle; out[4] = identity; out[5] = total;
    }
}

extern "C" void kernel_launch(void* const* d_in, const int* in_sizes, int n_in,
                              void* d_out, int out_size, void* d_ws, size_t ws_size,
                              hipStream_t stream) {
    const float* pred_noise   = (const float*)d_in[0];
    const float* target_noise = (const float*)d_in[1];
    const float* style_sim    = (const float*)d_in[2];
    const float* style_real   = (const float*)d_in[3];
    const float* content_orig = (const float*)d_in[4];
    const float* content_gen  = (const float*)d_in[5];
    const float* x_cycle      = (const float*)d_in[6];
    const float* x_original   = (const float*)d_in[7];
    const float* x_identity   = (const float*)d_in[8];

    float* ws  = (float*)d_ws;
    float* out = (float*)d_out;

    mse4_partial_kernel<<<4 * MSE_BLOCKS, MSE_THREADS, 0, stream>>>(
        pred_noise, target_noise,
        style_sim, style_real,
        content_orig, content_gen,
        x_original, x_identity,
        ws);

    chamfer_both_kernel<<<2 * CHAMFER_GRID, 256, 0, stream>>>(
        x_original, x_cycle, ws + 256);

    finalize_kernel<<<1, 256, 0, stream>>>(ws, out);
}